// GNN_45707041964408
// MI455X (gfx1250) — compile-verified
//
#include <hip/hip_runtime.h>
#include <hip/hip_bf16.h>
#include <cstdint>

// ---------------------------------------------------------------------------
// GAT (2-layer, 4 heads) for MI455X / gfx1250, wave32.
//   Layer GEMMs use V_WMMA_F32_16X16X4_F32 (full fp32 matrix pipe).
//   Weights staged to LDS with GLOBAL_LOAD_ASYNC_TO_LDS_B128 when available.
//   Edge softmax uses ordered-uint atomicMax + f32 atomicAdd.
// ---------------------------------------------------------------------------

#define USE_ASYNC_STAGE 1

typedef float v2f __attribute__((ext_vector_type(2)));
typedef float v8f __attribute__((ext_vector_type(8)));
typedef int   gv4i __attribute__((vector_size(16)));   // clang builtin 'V4i' kind

#define NEG_SLOPE 0.2f

__device__ __forceinline__ float leaky(float x) { return x >= 0.f ? x : NEG_SLOPE * x; }

// Monotonic float<->uint mapping so unsigned atomicMax implements float max.
__device__ __forceinline__ unsigned f2ord(float f) {
  unsigned u = __float_as_uint(f);
  return (u & 0x80000000u) ? ~u : (u | 0x80000000u);
}
__device__ __forceinline__ float ord2f(unsigned u) {
  return __uint_as_float((u & 0x80000000u) ? (u ^ 0x80000000u) : ~u);
}

// ---------------------------------------------------------------------------
// Row-tiled fp32 WMMA GEMM:  D[N x NC] = A[N x K] @ W[K x NC]
//   - fully compile-time tiling (no dynamic VGPR indexing)
//   - W staged once per block in LDS (async bulk copy when the toolchain has
//     the gfx1250 async-to-LDS builtin)
//   - ROWT row tiles per wave reuse each B fragment
// WMMA 16x16x4 f32 operand mapping (wave32):
//   A: lanes 0-15 rows M, laneHi selects K offset {0,2}; vgpr selects +1.
//   B: lanes 0-15 cols N, same K interleave, sourced from LDS.
//   C/D: vgpr r holds row r (+8 for lanes 16-31), lane%16 = col.
// ---------------------------------------------------------------------------
template <int K, int NC, int ROWT>
__global__ void __launch_bounds__(256)
gat_gemm_wmma(const float* __restrict__ A, const float* __restrict__ W,
              float* __restrict__ D, int Nrows) {
  constexpr int NCT = NC / 16;                 // column tiles (1 or 2)
  __shared__ __align__(16) float ldsW[K * NC]; // straight row-major copy of W

  const int tid = threadIdx.x;
#if USE_ASYNC_STAGE && __has_builtin(__builtin_amdgcn_global_load_async_to_lds_b128)
  {
    const uintptr_t gbase = (uintptr_t)W;
    const uintptr_t lbase = (uintptr_t)&ldsW[0];   // flat LDS addr: low 32b = offset
    for (int byteOff = tid * 16; byteOff < K * NC * 4; byteOff += blockDim.x * 16) {
      __builtin_amdgcn_global_load_async_to_lds_b128(
          (__attribute__((address_space(1))) gv4i*)(gbase + byteOff),
          (__attribute__((address_space(3))) gv4i*)(uint32_t)(lbase + byteOff),
          0, 0);
    }
#if __has_builtin(__builtin_amdgcn_s_wait_asynccnt)
    __builtin_amdgcn_s_wait_asynccnt(0);
#else
    asm volatile("s_wait_asynccnt 0" ::: "memory");
#endif
  }
#else
  for (int p = tid * 4; p < K * NC; p += blockDim.x * 4)
    *(float4*)(ldsW + p) = *(const float4*)(W + p);
#endif
  __syncthreads();

  const int wave   = tid >> 5;
  const int lane   = tid & 31;
  const int laneM  = lane & 15;
  const int laneHi = lane >> 4;                // 0: K+{0,1}, 1: K+{2,3}
  const int nWaves = blockDim.x >> 5;
  constexpr int ROWS = 16 * ROWT;
  const int nTiles = (Nrows + ROWS - 1) / ROWS;
  const int tStride = gridDim.x * nWaves;

  for (int t = blockIdx.x * nWaves + wave; t < nTiles; t += tStride) {
    const int row0 = t * ROWS;
    const float* arow[ROWT];
#pragma unroll
    for (int rt = 0; rt < ROWT; ++rt) {
      int r = row0 + rt * 16 + laneM;
      if (r >= Nrows) r = Nrows - 1;           // clamp (stores are guarded)
      arow[rt] = A + (size_t)r * K;
    }

    // Prefetch next grid-stride tile's row heads while this tile computes.
    const int tnext = t + tStride;
    if (tnext < nTiles) {
#pragma unroll
      for (int rt = 0; rt < ROWT; ++rt)
        __builtin_prefetch(A + (size_t)(tnext * ROWS + rt * 16 + laneM) * K, 0, 3);
    }

    v8f acc[ROWT][NCT];
#pragma unroll
    for (int rt = 0; rt < ROWT; ++rt)
#pragma unroll
      for (int ct = 0; ct < NCT; ++ct)
#pragma unroll
        for (int j = 0; j < 8; ++j) acc[rt][ct][j] = 0.f;

#pragma unroll 4
    for (int k = 0; k < K; k += 4) {
      const int ka = k + laneHi * 2;
      v2f b[NCT];
#pragma unroll
      for (int ct = 0; ct < NCT; ++ct) {
        const int col = ct * 16 + laneM;
        b[ct][0] = ldsW[(size_t)ka * NC + col];
        b[ct][1] = ldsW[(size_t)(ka + 1) * NC + col];
      }
#pragma unroll
      for (int rt = 0; rt < ROWT; ++rt) {
        const v2f a = *(const v2f*)(arow[rt] + ka);
#pragma unroll
        for (int ct = 0; ct < NCT; ++ct)
          acc[rt][ct] = __builtin_amdgcn_wmma_f32_16x16x4_f32(
              false, a, false, b[ct], (short)0, acc[rt][ct], false, false);
      }
    }

    if (row0 + ROWS <= Nrows) {
      // Fast path: full tile, no per-element guards / exec-mask juggling.
#pragma unroll
      for (int rt = 0; rt < ROWT; ++rt)
#pragma unroll
        for (int ct = 0; ct < NCT; ++ct) {
          float* dcol = D + (size_t)(row0 + rt * 16 + laneHi * 8) * NC + ct * 16 + laneM;
#pragma unroll
          for (int r = 0; r < 8; ++r) dcol[(size_t)r * NC] = acc[rt][ct][r];
        }
    } else {
#pragma unroll
      for (int rt = 0; rt < ROWT; ++rt)
#pragma unroll
        for (int ct = 0; ct < NCT; ++ct) {
          const int col = ct * 16 + laneM;
#pragma unroll
          for (int r = 0; r < 8; ++r) {
            const int row = row0 + rt * 16 + r + laneHi * 8;
            if (row < Nrows) D[(size_t)row * NC + col] = acc[rt][ct][r];
          }
        }
    }
  }
}

// e_src[n][h] = dot(h[n,h,:], a_src[h,:]) ; likewise e_dst. One thread per (n,h).
template <int C>
__global__ void gat_att_scores(const float* __restrict__ h,
                               const float* __restrict__ a_src,
                               const float* __restrict__ a_dst,
                               float* __restrict__ es, float* __restrict__ ed, int N) {
  const int i = blockIdx.x * blockDim.x + threadIdx.x;
  if (i >= N * 4) return;
  const int n = i >> 2, hh = i & 3;
  const float* __restrict__ hp = h + (size_t)n * 4 * C + hh * C;
  const float* __restrict__ as = a_src + hh * C;
  const float* __restrict__ ad = a_dst + hh * C;
  float s = 0.f, d = 0.f;
#pragma unroll
  for (int c = 0; c < C; ++c) { s += hp[c] * as[c]; d += hp[c] * ad[c]; }
  es[i] = s;
  ed[i] = d;
}

// Pass 1: segment max of leaky(e_src[src]+e_dst[dst]) by dst, per head.
__global__ void gat_edge_max(const int* __restrict__ ei, int E, int N,
                             const float* __restrict__ es, const float* __restrict__ ed,
                             unsigned* __restrict__ m) {
  const int i = blockIdx.x * blockDim.x + threadIdx.x;
  if (i >= E + N) return;
  const int s = (i < E) ? ei[i] : (i - E);          // implicit self-loops
  const int d = (i < E) ? ei[E + i] : (i - E);
  const float4 a = *(const float4*)(es + (size_t)s * 4);
  const float4 b = *(const float4*)(ed + (size_t)d * 4);
  unsigned* md = m + (size_t)d * 4;
  atomicMax(md + 0, f2ord(leaky(a.x + b.x)));
  atomicMax(md + 1, f2ord(leaky(a.y + b.y)));
  atomicMax(md + 2, f2ord(leaky(a.z + b.z)));
  atomicMax(md + 3, f2ord(leaky(a.w + b.w)));
}

// Pass 2: denom[dst][h] += exp(e - m[dst][h])
__global__ void gat_edge_sum(const int* __restrict__ ei, int E, int N,
                             const float* __restrict__ es, const float* __restrict__ ed,
                             const unsigned* __restrict__ m, float* __restrict__ den) {
  const int i = blockIdx.x * blockDim.x + threadIdx.x;
  if (i >= E + N) return;
  const int s = (i < E) ? ei[i] : (i - E);
  const int d = (i < E) ? ei[E + i] : (i - E);
  const float4 a = *(const float4*)(es + (size_t)s * 4);
  const float4 b = *(const float4*)(ed + (size_t)d * 4);
  const unsigned* md = m + (size_t)d * 4;
  float* dd = den + (size_t)d * 4;
  atomicAdd(dd + 0, __expf(leaky(a.x + b.x) - ord2f(md[0])));
  atomicAdd(dd + 1, __expf(leaky(a.y + b.y) - ord2f(md[1])));
  atomicAdd(dd + 2, __expf(leaky(a.z + b.z) - ord2f(md[2])));
  atomicAdd(dd + 3, __expf(leaky(a.w + b.w) - ord2f(md[3])));
}

// Pass 3: out[dst] += h[src] * alpha  (alpha recomputed from m, den)
template <int C>
__global__ void gat_edge_agg(const int* __restrict__ ei, int E, int N,
                             const float* __restrict__ es, const float* __restrict__ ed,
                             const float* __restrict__ hfeat,
                             const unsigned* __restrict__ m, const float* __restrict__ den,
                             float* __restrict__ out) {
  const int i = blockIdx.x * blockDim.x + threadIdx.x;
  if (i >= E + N) return;
  const int s = (i < E) ? ei[i] : (i - E);
  const int d = (i < E) ? ei[E + i] : (i - E);
  const float4 a = *(const float4*)(es + (size_t)s * 4);
  const float4 b = *(const float4*)(ed + (size_t)d * 4);
  const float ea[4] = {a.x + b.x, a.y + b.y, a.z + b.z, a.w + b.w};
  const unsigned* md = m + (size_t)d * 4;
  const float* dd = den + (size_t)d * 4;
  const float* __restrict__ hs = hfeat + (size_t)s * 4 * C;
  float* __restrict__ od = out + (size_t)d * 4 * C;
#pragma unroll
  for (int hh = 0; hh < 4; ++hh) {
    const float alpha = __expf(leaky(ea[hh]) - ord2f(md[hh])) / (dd[hh] + 1e-16f);
#pragma unroll
    for (int c = 0; c < C; ++c)
      atomicAdd(od + hh * C + c, hs[hh * C + c] * alpha);
  }
}

__global__ void gat_bias_relu(float* __restrict__ y, const float* __restrict__ b,
                              int total, int mask) {
  const int i = blockIdx.x * blockDim.x + threadIdx.x;
  if (i >= total) return;
  const float v = y[i] + b[i & mask];
  y[i] = v > 0.f ? v : 0.f;
}

__global__ void gat_pool(const float* __restrict__ y, const int* __restrict__ batch,
                         float* __restrict__ pooled, float* __restrict__ cnt, int N) {
  const int n = blockIdx.x * blockDim.x + threadIdx.x;
  if (n >= N) return;
  const int g = batch[n];
  atomicAdd(&cnt[g], 1.0f);
  const float* yn = y + (size_t)n * 16;
#pragma unroll
  for (int c = 0; c < 16; ++c) atomicAdd(&pooled[(size_t)g * 16 + c], yn[c]);
}

__global__ void gat_final(const float* __restrict__ pooled, const float* __restrict__ cnt,
                          const float* __restrict__ Wf, const float* __restrict__ bf,
                          float* __restrict__ out, int G) {
  const int i = blockIdx.x * blockDim.x + threadIdx.x;
  if (i >= G * 10) return;
  const int g = i / 10, o = i % 10;
  const float inv = 1.f / fmaxf(cnt[g], 1.f);
  float acc = bf[o];
#pragma unroll
  for (int c = 0; c < 16; ++c) acc += pooled[(size_t)g * 16 + c] * inv * Wf[c * 10 + o];
  out[i] = acc;
}

__global__ void gat_zero(float* __restrict__ p, size_t n) {
  size_t i = blockIdx.x * (size_t)blockDim.x + threadIdx.x;
  const size_t stride = (size_t)gridDim.x * blockDim.x;
  for (; i < n; i += stride) p[i] = 0.f;
}

// ---------------------------------------------------------------------------
extern "C" void kernel_launch(void* const* d_in, const int* in_sizes, int n_in,
                              void* d_out, int out_size, void* d_ws, size_t ws_size,
                              hipStream_t stream) {
  const float* x    = (const float*)d_in[0];
  const int*   ei   = (const int*)d_in[1];    // [2,E] row-major: src then dst
  const int*   bat  = (const int*)d_in[2];
  const float* W1   = (const float*)d_in[3];
  const float* as1  = (const float*)d_in[4];
  const float* ad1  = (const float*)d_in[5];
  const float* b1   = (const float*)d_in[6];
  const float* W2   = (const float*)d_in[7];
  const float* as2  = (const float*)d_in[8];
  const float* ad2  = (const float*)d_in[9];
  const float* b2   = (const float*)d_in[10];
  const float* Wf   = (const float*)d_in[11];
  const float* bf   = (const float*)d_in[12];

  const int N = in_sizes[0] / 512;
  const int E = in_sizes[1] / 2;
  const int G = out_size / 10;

  // Workspace layout (floats).
  float* ws = (float*)d_ws;
  size_t off = 0;
  float*    h1   = ws + off; off += (size_t)N * 32;
  float*    es1  = ws + off; off += (size_t)N * 4;
  float*    ed1  = ws + off; off += (size_t)N * 4;
  float*    h2   = ws + off; off += (size_t)N * 16;
  float*    es2  = ws + off; off += (size_t)N * 4;
  float*    ed2  = ws + off; off += (size_t)N * 4;
  float*    accBase = ws + off;                       // everything below is zeroed
  unsigned* m1   = (unsigned*)(ws + off); off += (size_t)N * 4;
  float*    den1 = ws + off; off += (size_t)N * 4;
  float*    out1 = ws + off; off += (size_t)N * 32;
  unsigned* m2   = (unsigned*)(ws + off); off += (size_t)N * 4;
  float*    den2 = ws + off; off += (size_t)N * 4;
  float*    out2 = ws + off; off += (size_t)N * 16;
  float*    pooled = ws + off; off += (size_t)G * 16;
  float*    cnt  = ws + off; off += (size_t)G;
  const size_t accFloats = (size_t)N * (4 + 4 + 32 + 4 + 4 + 16) + (size_t)G * 17;
  (void)ws_size; (void)n_in;

  const int ET = E + N;
  const int eBlocks = (ET + 255) / 256;
  const int nTiles32 = (N + 31) / 32;                 // 32 rows per wave (ROWT=2)
  int gemmBlocks = (nTiles32 + 7) / 8;                // 8 waves/block, grid-stride
  if (gemmBlocks > 1024) gemmBlocks = 1024;

  // 0) zero accumulators (m=0 is the ordered-uint identity for float max)
  gat_zero<<<2048, 256, 0, stream>>>(accBase, accFloats);

  // ---- Layer 1 ----
  gat_gemm_wmma<512, 32, 2><<<gemmBlocks, 256, 0, stream>>>(x, W1, h1, N);
  gat_att_scores<8><<<(N * 4 + 255) / 256, 256, 0, stream>>>(h1, as1, ad1, es1, ed1, N);
  gat_edge_max<<<eBlocks, 256, 0, stream>>>(ei, E, N, es1, ed1, m1);
  gat_edge_sum<<<eBlocks, 256, 0, stream>>>(ei, E, N, es1, ed1, m1, den1);
  gat_edge_agg<8><<<eBlocks, 256, 0, stream>>>(ei, E, N, es1, ed1, h1, m1, den1, out1);
  gat_bias_relu<<<(N * 32 + 255) / 256, 256, 0, stream>>>(out1, b1, N * 32, 31);

  // ---- Layer 2 ----
  gat_gemm_wmma<32, 16, 2><<<gemmBlocks, 256, 0, stream>>>(out1, W2, h2, N);
  gat_att_scores<4><<<(N * 4 + 255) / 256, 256, 0, stream>>>(h2, as2, ad2, es2, ed2, N);
  gat_edge_max<<<eBlocks, 256, 0, stream>>>(ei, E, N, es2, ed2, m2);
  gat_edge_sum<<<eBlocks, 256, 0, stream>>>(ei, E, N, es2, ed2, m2, den2);
  gat_edge_agg<4><<<eBlocks, 256, 0, stream>>>(ei, E, N, es2, ed2, h2, m2, den2, out2);
  gat_bias_relu<<<(N * 16 + 255) / 256, 256, 0, stream>>>(out2, b2, N * 16, 15);

  // ---- Pool + classifier ----
  gat_pool<<<(N + 255) / 256, 256, 0, stream>>>(out2, bat, pooled, cnt, N);
  gat_final<<<(G * 10 + 255) / 256, 256, 0, stream>>>(pooled, cnt, Wf, bf, (float*)d_out, G);
}